// learnable_computation_tree_22196390986046
// MI455X (gfx1250) — compile-verified
//
#include <hip/hip_runtime.h>
#include <hip/hip_bf16.h>

typedef float v2f __attribute__((ext_vector_type(2)));
typedef float v8f __attribute__((ext_vector_type(8)));

// ---- FEX operator pools -----------------------------------------------------
// UNARY = [id, sin, cos, square, tanh], BINARY = [add, mul]
__device__ __forceinline__ float unary_apply(int act, float z) {
    switch (act) {
        case 1:  return __sinf(z);
        case 2:  return __cosf(z);
        case 3:  return z * z;
        case 4:  return tanhf(z);
        default: return z;
    }
}

// Whole-tile unary: ONE uniform branch per leaf per tile (not per element),
// so each case's 8 ops schedule together under a single s_cbranch diamond.
__device__ __forceinline__ void unary_tile(int act, const v2f a[4], v2f o[4]) {
    switch (act) {
        case 1:
#pragma unroll
            for (int c = 0; c < 4; ++c) { o[c].x = __sinf(a[c].x); o[c].y = __sinf(a[c].y); }
            break;
        case 2:
#pragma unroll
            for (int c = 0; c < 4; ++c) { o[c].x = __cosf(a[c].x); o[c].y = __cosf(a[c].y); }
            break;
        case 3:
#pragma unroll
            for (int c = 0; c < 4; ++c) { o[c].x = a[c].x * a[c].x; o[c].y = a[c].y * a[c].y; }
            break;
        case 4:
#pragma unroll
            for (int c = 0; c < 4; ++c) { o[c].x = tanhf(a[c].x); o[c].y = tanhf(a[c].y); }
            break;
        default:
#pragma unroll
            for (int c = 0; c < 4; ++c) o[c] = a[c];
            break;
    }
}

__device__ __forceinline__ float binary_apply(int act, float u, float v) {
    return (act == 1) ? (u * v) : (u + v);
}

// ---- Kernel -----------------------------------------------------------------
// One wave32 handles a 16-row tile:
//   D(16x16,f32) += A_leaf(16x4,f32) x B_leaf(4x16,f32)   via v_wmma_f32_16x16x4_f32
// B for leaf i has W[i] in column i only, so one accumulator collects all three
// leaf dot-products (columns 0..2 of D = v0, v2, v4).
__global__ __launch_bounds__(256) void fex_tree_wmma_kernel(
    const float* __restrict__ x,      // [n,16]
    const int*   __restrict__ bs,     // [6]
    const float* __restrict__ W,      // [3,16]
    const float* __restrict__ bvec,   // [3]
    const float* __restrict__ a5p,    // [1]
    const float* __restrict__ b5p,    // [1]
    float* __restrict__ out,          // [n]
    int n)
{
    __shared__ float xch[8][3][16];   // [wave][leaf-column][row-in-tile]

    const int lane = threadIdx.x & 31;
    const int wid  = threadIdx.x >> 5;
    const int col  = lane & 15;       // matrix N-index / row M-index role
    const int hi   = lane >> 4;       // 0: K offset 0, 1: K offset +2 (f32 WMMA layout)

    // Uniform (scalar) tree parameters.
    const int act0 = bs[0], act1 = bs[1], act2 = bs[2];
    const int act3 = bs[3], act4 = bs[4], act5 = bs[5];
    const float a5 = a5p[0], b5 = b5p[0];
    const float bb0 = bvec[0], bb1 = bvec[1], bb2 = bvec[2];

    // W into SGPRs (uniform addresses -> s_load).
    float wv[3][16];
#pragma unroll
    for (int i = 0; i < 3; ++i)
#pragma unroll
        for (int k = 0; k < 16; ++k)
            wv[i][k] = W[i * 16 + k];

    // Hoisted B operands: leaf i occupies column i only.
    // B chunk c covers K = 4c..4c+3; VGPR0 holds K=4c+2*hi, VGPR1 holds K=4c+1+2*hi.
    v2f Bm[3][4];
#pragma unroll
    for (int i = 0; i < 3; ++i) {
#pragma unroll
        for (int c = 0; c < 4; ++c) {
            float lo = hi ? wv[i][4 * c + 2] : wv[i][4 * c + 0];
            float hh = hi ? wv[i][4 * c + 3] : wv[i][4 * c + 1];
            v2f bv; bv.x = lo; bv.y = hh;
            v2f zz; zz.x = 0.0f; zz.y = 0.0f;
            Bm[i][c] = (col == i) ? bv : zz;
        }
    }

    const int tiles   = (n + 15) >> 4;
    const int wstride = gridDim.x * (blockDim.x >> 5);

    for (int t = blockIdx.x * 8 + wid; t < tiles; t += wstride) {
        const int row_base = t << 4;
        int r = row_base + col;
        if (r >= n) r = n - 1;                  // clamp loads; stores are guarded
        const v2f* p = (const v2f*)(x + (size_t)r * 16);

        // Prefetch next grid-stride tile (each row = exactly one 64B cacheline).
        {
            int tn = t + wstride;
            if (tn < tiles) {
                int rn = (tn << 4) + col;
                if (rn >= n) rn = n - 1;
                __builtin_prefetch(x + (size_t)rn * 16, 0, 0);
            }
        }

        // A tile: lane owns row (col); hi selects K parity per the f32 A layout.
        v2f a[4];
#pragma unroll
        for (int c = 0; c < 4; ++c)
            a[c] = __builtin_nontemporal_load(p + 2 * c + hi);

        // Per-leaf unary transform: one uniform branch per leaf.
        v2f A0[4], A2[4], A4[4];
        unary_tile(act0, a, A0);
        unary_tile(act2, a, A2);
        unary_tile(act4, a, A4);

        // Three leaves accumulate into one D.
        v8f acc = {};
#pragma unroll
        for (int c = 0; c < 4; ++c)
            acc = __builtin_amdgcn_wmma_f32_16x16x4_f32(
                false, A0[c], false, Bm[0][c], (short)0, acc, false, false);
#pragma unroll
        for (int c = 0; c < 4; ++c)
            acc = __builtin_amdgcn_wmma_f32_16x16x4_f32(
                false, A2[c], false, Bm[1][c], (short)0, acc, false, false);
#pragma unroll
        for (int c = 0; c < 4; ++c)
            acc = __builtin_amdgcn_wmma_f32_16x16x4_f32(
                false, A4[c], false, Bm[2][c], (short)0, acc, false, false);

        // D layout: VGPR v, lanes 0-15 -> M=v, lanes 16-31 -> M=v+8.
        // Lanes holding columns 0..2 spill their 8 row-values to LDS.
        if (col < 3) {
            float* dst = &xch[wid][col][hi * 8];
#pragma unroll
            for (int v = 0; v < 8; ++v)
                dst[v] = acc[v];
        }
        asm volatile("s_wait_dscnt 0" ::: "memory");

        // Lanes 0-15: per-row scalar combine tree + store.
        if (hi == 0) {
            float v0 = xch[wid][0][col] + bb0;
            float v2 = xch[wid][1][col] + bb1;
            float v4 = xch[wid][2][col] + bb2;
            float left  = binary_apply(act1, v0, v2);
            float right = a5 * unary_apply(act5, v4) + b5;
            float o     = binary_apply(act3, left, right);
            int orow = row_base + col;
            if (orow < n)
                __builtin_nontemporal_store(o, out + orow);
        }
        asm volatile("" ::: "memory");   // keep next tile's LDS stores ordered after reads
    }
}

// ---- Launch -----------------------------------------------------------------
extern "C" void kernel_launch(void* const* d_in, const int* in_sizes, int n_in,
                              void* d_out, int out_size, void* d_ws, size_t ws_size,
                              hipStream_t stream) {
    const float* x    = (const float*)d_in[0];
    const int*   bs   = (const int*)d_in[1];
    const float* W    = (const float*)d_in[2];
    const float* bvec = (const float*)d_in[3];
    const float* a5   = (const float*)d_in[4];
    const float* b5   = (const float*)d_in[5];
    float* out = (float*)d_out;

    const int n     = in_sizes[0] / 16;       // rows
    const int tiles = (n + 15) / 16;
    int blocks = (tiles + 7) / 8;             // 8 waves (tiles) per 256-thread block
    if (blocks > 3072) blocks = 3072;         // grid-stride to amortize W/B setup
    if (blocks < 1)    blocks = 1;

    fex_tree_wmma_kernel<<<blocks, 256, 0, stream>>>(x, bs, W, bvec, a5, b5, out, n);
}